// NodeNetwork_3255585210371
// MI455X (gfx1250) — compile-verified
//
#include <hip/hip_runtime.h>
#include <hip/hip_bf16.h>

typedef float v2f __attribute__((ext_vector_type(2)));
typedef float v8f __attribute__((ext_vector_type(8)));

#define W1_STR 130   // padded LDS stride for W1^T rows (even -> 8B aligned float2)
#define W2_STR 66    // padded LDS stride for W2^T / W3^T rows
#define HS_STR 66    // padded LDS stride for per-wave activation tile rows

static __device__ __forceinline__ v8f v8f_zero() {
  v8f z = {0.f, 0.f, 0.f, 0.f, 0.f, 0.f, 0.f, 0.f};
  return z;
}

// ---------------- scatter phase ----------------

__global__ __launch_bounds__(256) void nn_zero(float4* __restrict__ p, int n4) {
  int i = blockIdx.x * blockDim.x + threadIdx.x;
  if (i < n4) p[i] = make_float4(0.f, 0.f, 0.f, 0.f);
}

// 16 lanes per edge, 4 features per lane: float4 gathers + 8 f32 atomics.
__global__ __launch_bounds__(256) void nn_edge_scatter(
    const float* __restrict__ x, const float* __restrict__ ew,
    const int* __restrict__ es, const int* __restrict__ ed,
    float* __restrict__ mi, int nE) {
  int gid = blockIdx.x * 256 + threadIdx.x;
  int eid = gid >> 4;
  if (eid >= nE) return;
  int f = (gid & 15) << 2;
  int s = es[eid];
  int t = ed[eid];
  float w = ew[eid];
  float4 xs = *(const float4*)(x + s * 64 + f);
  float4 xt = *(const float4*)(x + t * 64 + f);
  float* mt = mi + t * 64 + f;
  float* ms = mi + s * 64 + f;
  unsafeAtomicAdd(mt + 0, w * xs.x);
  unsafeAtomicAdd(mt + 1, w * xs.y);
  unsafeAtomicAdd(mt + 2, w * xs.z);
  unsafeAtomicAdd(mt + 3, w * xs.w);
  unsafeAtomicAdd(ms + 0, w * xt.x);
  unsafeAtomicAdd(ms + 1, w * xt.y);
  unsafeAtomicAdd(ms + 2, w * xt.z);
  unsafeAtomicAdd(ms + 3, w * xt.w);
}

// ---------------- fused MLP phase (WMMA f32 16x16x4) ----------------

// D-layout layernorm + tanh + store to per-wave LDS tile.
// acc[t][v] holds element (row m0+v+8*lhi, col t*16+lmod) of the 16x64 tile.
static __device__ __forceinline__ void ln_tanh_store(
    v8f acc[4], const float* __restrict__ bias, const float* __restrict__ gain,
    const float* __restrict__ beta, float* hs, int lmod, int lhi) {
  float bn[4], gn[4], en[4];
#pragma unroll
  for (int t = 0; t < 4; ++t) {
    int n = t * 16 + lmod;
    bn[t] = bias[n];
    gn[t] = gain[n];
    en[t] = beta[n];
  }
#pragma unroll
  for (int v = 0; v < 8; ++v) {
    float s = 0.f, q = 0.f;
#pragma unroll
    for (int t = 0; t < 4; ++t) {
      float hv = acc[t][v] + bn[t];
      acc[t][v] = hv;
      s += hv;
      q += hv * hv;
    }
    // butterfly over 16-lane halves (masks never cross the half boundary)
#pragma unroll
    for (int m = 1; m <= 8; m <<= 1) {
      s += __shfl_xor(s, m, 32);
      q += __shfl_xor(q, m, 32);
    }
    float mu = s * 0.015625f;                          // /64
    float inv = rsqrtf(q * 0.015625f - mu * mu + 1e-5f);
    int row = (v + 8 * lhi) * HS_STR;
#pragma unroll
    for (int t = 0; t < 4; ++t) {
      float hv = (acc[t][v] - mu) * inv * gn[t] + en[t];
      hs[row + t * 16 + lmod] = tanhf(hv);
    }
  }
}

// 16x64 += (16x64 from LDS) @ (64x64 W^T from LDS), K-steps of 4 via WMMA.
static __device__ __forceinline__ void gemm64_lds(v8f acc[4], const float* hs,
                                                  const float* Wt, int lmod,
                                                  int lhi) {
#pragma unroll
  for (int kk = 0; kk < 16; ++kk) {
    int k = kk * 4 + lhi * 2;
    v2f a = *(const v2f*)(hs + lmod * HS_STR + k);
#pragma unroll
    for (int t = 0; t < 4; ++t) {
      v2f b = *(const v2f*)(Wt + (t * 16 + lmod) * W2_STR + k);
      acc[t] = __builtin_amdgcn_wmma_f32_16x16x4_f32(false, a, false, b,
                                                     (short)0, acc[t], false,
                                                     false);
    }
  }
}

__global__ __launch_bounds__(256) void nn_mlp(
    const float* __restrict__ mi, const float* __restrict__ x,
    const float* __restrict__ W1, const float* __restrict__ b1,
    const float* __restrict__ g1, const float* __restrict__ be1,
    const float* __restrict__ W2, const float* __restrict__ b2,
    const float* __restrict__ g2, const float* __restrict__ be2,
    const float* __restrict__ W3, const float* __restrict__ b3,
    float* __restrict__ out, int nN) {
  __shared__ float W1t[64 * W1_STR];   // W1^T [n][k], k=0..127
  __shared__ float W2t[64 * W2_STR];   // W2^T [n][k], k=0..63
  __shared__ float W3t[64 * W2_STR];   // W3^T
  __shared__ float HS[8 * 16 * HS_STR];// per-wave 16x64 staging tiles

  // coalesced global reads, padded transposed LDS writes
  for (int idx = threadIdx.x; idx < 128 * 64; idx += 256) {
    int k = idx >> 6, n = idx & 63;
    W1t[n * W1_STR + k] = W1[idx];
  }
  for (int idx = threadIdx.x; idx < 64 * 64; idx += 256) {
    int k = idx >> 6, n = idx & 63;
    W2t[n * W2_STR + k] = W2[idx];
    W3t[n * W2_STR + k] = W3[idx];
  }
  __syncthreads();

  int wave = threadIdx.x >> 5;
  int lane = threadIdx.x & 31;
  int tile = blockIdx.x * 8 + wave;
  int m0 = tile * 16;
  if (m0 >= nN) return;  // wave-uniform; EXEC stays all-1s for WMMA below
  int lmod = lane & 15;
  int lhi = lane >> 4;
  float* hs = HS + wave * (16 * HS_STR);

  v8f acc[4];
#pragma unroll
  for (int t = 0; t < 4; ++t) acc[t] = v8f_zero();

  // ---- layer 1: A = [mi | x] (16 x 128) ----
  int rowc = m0 + lmod;
  if (rowc >= nN) rowc = nN - 1;  // safety clamp (N is a multiple of 16 here)
  long rowbase = (long)rowc * 64;
#pragma unroll 4
  for (int kk = 0; kk < 32; ++kk) {
    int k = kk * 4 + lhi * 2;  // k and k+2 stay on the same side of 64
    v2f a = (k < 64) ? *(const v2f*)(mi + rowbase + k)
                     : *(const v2f*)(x + rowbase + (k - 64));
#pragma unroll
    for (int t = 0; t < 4; ++t) {
      v2f b = *(const v2f*)(W1t + (t * 16 + lmod) * W1_STR + k);
      acc[t] = __builtin_amdgcn_wmma_f32_16x16x4_f32(false, a, false, b,
                                                     (short)0, acc[t], false,
                                                     false);
    }
  }
  ln_tanh_store(acc, b1, g1, be1, hs, lmod, lhi);

  // ---- layer 2 ----
#pragma unroll
  for (int t = 0; t < 4; ++t) acc[t] = v8f_zero();
  gemm64_lds(acc, hs, W2t, lmod, lhi);
  ln_tanh_store(acc, b2, g2, be2, hs, lmod, lhi);

  // ---- layer 3 (no LN/tanh) ----
#pragma unroll
  for (int t = 0; t < 4; ++t) acc[t] = v8f_zero();
  gemm64_lds(acc, hs, W3t, lmod, lhi);

#pragma unroll
  for (int t = 0; t < 4; ++t) {
    float bn = b3[t * 16 + lmod];
#pragma unroll
    for (int v = 0; v < 8; ++v)
      hs[(v + 8 * lhi) * HS_STR + t * 16 + lmod] = acc[t][v] + bn;
  }

  // coalesced write-out: 32 lanes x float2 = 256B contiguous per iteration
#pragma unroll
  for (int i = 0; i < 16; ++i) {
    int flat = i * 32 + lane;  // float2 index in 16x64 tile
    int r = flat >> 5;
    int c2 = flat & 31;
    if (m0 + r < nN) {
      v2f val = *(const v2f*)(hs + r * HS_STR + c2 * 2);
      *(v2f*)(out + (long)(m0 + r) * 64 + c2 * 2) = val;
    }
  }
}

// ---------------- host launcher ----------------

extern "C" void kernel_launch(void* const* d_in, const int* in_sizes, int n_in,
                              void* d_out, int out_size, void* d_ws,
                              size_t ws_size, hipStream_t stream) {
  const float* x = (const float*)d_in[0];
  const float* e = (const float*)d_in[1];
  const int* ei = (const int*)d_in[2];  // (2,E): [start; end]
  const float* W1 = (const float*)d_in[3];
  const float* b1 = (const float*)d_in[4];
  const float* g1 = (const float*)d_in[5];
  const float* be1 = (const float*)d_in[6];
  const float* W2 = (const float*)d_in[7];
  const float* b2 = (const float*)d_in[8];
  const float* g2 = (const float*)d_in[9];
  const float* be2 = (const float*)d_in[10];
  const float* W3 = (const float*)d_in[11];
  const float* b3 = (const float*)d_in[12];
  float* out = (float*)d_out;

  int nE = in_sizes[1];
  int nN = in_sizes[0] / 64;
  float* mi = (float*)d_ws;  // N x 64 fp32 accumulator

  int n4 = nN * 16;  // nN*64/4 float4 elements
  nn_zero<<<(n4 + 255) / 256, 256, 0, stream>>>((float4*)mi, n4);

  long tE = (long)nE * 16;
  nn_edge_scatter<<<(int)((tE + 255) / 256), 256, 0, stream>>>(x, e, ei,
                                                               ei + nE, mi, nE);

  int tiles = (nN + 15) / 16;
  int blocks = (tiles + 7) / 8;
  nn_mlp<<<blocks, 256, 0, stream>>>(mi, x, W1, b1, g1, be1, W2, b2, g2, be2,
                                     W3, b3, out, nN);
}